// Model_65678639890810
// MI455X (gfx1250) — compile-verified
//
#include <hip/hip_runtime.h>
#include <hip/hip_bf16.h>
#include <stdint.h>

#define HWIN 2
#define TILE 32
#define PADW (TILE + 2 * HWIN)   // 36
#define IMG_H 512
#define IMG_W 512

__device__ __forceinline__ int iclamp(int v, int lo, int hi) {
    return v < lo ? lo : (v > hi ? hi : v);
}

__global__ __launch_bounds__(256) void bilateral_pass(
    const float* __restrict__ src, float* __restrict__ dst,
    const float* __restrict__ sigma_xyz, const float* __restrict__ sigma_r,
    int pass)
{
    __shared__ float tile[PADW * PADW];

    const int b    = blockIdx.z;
    const int row0 = blockIdx.y * TILE;
    const int col0 = blockIdx.x * TILE;
    const float* img = src + (size_t)b * IMG_H * IMG_W;

    // ---- async global -> LDS staging of 36x36 tile with replicate clamp ----
    // Per-lane addressing: VDST = LDS byte offset, VADDR = 64-bit global addr.
    // Tracked by ASYNCcnt; bypasses VGPR staging (CDNA5 async path).
    const unsigned lds_base = (unsigned)(uintptr_t)&tile[0]; // flat low 32b = LDS offset
    const int tid = threadIdx.x;
    #pragma unroll
    for (int k = 0; k < 6; ++k) {
        int idx = tid + k * 256;
        if (idx < PADW * PADW) {
            int lr = idx / PADW;
            int lc = idx - lr * PADW;
            int gr = iclamp(row0 - HWIN + lr, 0, IMG_H - 1);
            int gc = iclamp(col0 - HWIN + lc, 0, IMG_W - 1);
            const float* gp = img + gr * IMG_W + gc;
            unsigned lds_addr = lds_base + (unsigned)idx * 4u;
            asm volatile("global_load_async_to_lds_b32 %0, %1, off"
                         :: "v"(lds_addr), "v"(gp)
                         : "memory");
        }
    }
    asm volatile("s_wait_asynccnt 0" ::: "memory");
    __syncthreads();

    // ---- per-pass sigmas (uniform scalar loads) ----
    const float sx = sigma_xyz[2 * pass + 0];   // row-offset sigma
    const float sy = sigma_xyz[2 * pass + 1];   // col-offset sigma
    const float sr = sigma_r[pass];
    const float LOG2E = 1.4426950408889634f;
    const float cx = (0.5f / (sx * sx)) * LOG2E;
    const float cy = (0.5f / (sy * sy)) * LOG2E;
    const float ncr = -(0.5f / (sr * sr)) * LOG2E;   // negative range coeff

    // Spatial weights folded into the exp2 exponent (log2 domain):
    // w = exp2( lsp[i][j] + ncr*d^2 ).  lsp is uniform -> lives in SGPRs.
    float lsp[5][5];
    #pragma unroll
    for (int i = 0; i < 5; ++i) {
        float ri = (float)((i - HWIN) * (i - HWIN)) * cx;
        #pragma unroll
        for (int j = 0; j < 5; ++j) {
            float rj = (float)((j - HWIN) * (j - HWIN)) * cy;
            lsp[i][j] = -(ri + rj);
        }
    }

    // ---- each thread: 2x2 pixel block from a 6x6 register window ----
    const int tcx = tid & 15;          // 0..15
    const int tcy = tid >> 4;          // 0..15
    const int pr  = tcy * 2;
    const int pc  = tcx * 2;

    float win[6][6];
    #pragma unroll
    for (int i = 0; i < 6; ++i)
        #pragma unroll
        for (int j = 0; j < 6; ++j)
            win[i][j] = tile[(pr + i) * PADW + (pc + j)];

    #pragma unroll
    for (int py = 0; py < 2; ++py) {
        #pragma unroll
        for (int px = 0; px < 2; ++px) {
            const float ctr = win[HWIN + py][HWIN + px];
            float num = 0.0f, den = 0.0f;
            #pragma unroll
            for (int i = 0; i < 5; ++i) {
                #pragma unroll
                for (int j = 0; j < 5; ++j) {
                    float nb = win[py + i][px + j];
                    float d  = nb - ctr;
                    float e  = fmaf(d * d, ncr, lsp[i][j]);
                    float w  = __builtin_amdgcn_exp2f(e);
                    num = fmaf(w, nb, num);
                    den += w;
                }
            }
            dst[((size_t)b * IMG_H + (row0 + pr + py)) * IMG_W + (col0 + pc + px)]
                = num * __builtin_amdgcn_rcpf(den);
        }
    }
}

extern "C" void kernel_launch(void* const* d_in, const int* in_sizes, int n_in,
                              void* d_out, int out_size, void* d_ws, size_t ws_size,
                              hipStream_t stream) {
    const float* x         = (const float*)d_in[0];  // [32,1,512,512]
    const float* sigma_xyz = (const float*)d_in[1];  // [3,2]
    const float* sigma_r   = (const float*)d_in[2];  // [3]
    float* out = (float*)d_out;
    float* ws  = (float*)d_ws;                        // needs 32 MiB

    dim3 grid(IMG_W / TILE, IMG_H / TILE, 32);
    dim3 block(256);

    // pass 0: x -> out, pass 1: out -> ws, pass 2: ws -> out
    bilateral_pass<<<grid, block, 0, stream>>>(x,   out, sigma_xyz, sigma_r, 0);
    bilateral_pass<<<grid, block, 0, stream>>>(out, ws,  sigma_xyz, sigma_r, 1);
    bilateral_pass<<<grid, block, 0, stream>>>(ws,  out, sigma_xyz, sigma_r, 2);
}